// AfmoeMoE_47665547051636
// MI455X (gfx1250) — compile-verified
//
#include <hip/hip_runtime.h>
#include <hip/hip_bf16.h>

// ---------------------------------------------------------------------------
// AFMoE: top-4 of 16 experts + shared expert, FP32 via V_WMMA_F32_16X16X4_F32
// M_TILE = 32 gathered rows per block; async global->LDS gather (ASYNCcnt).
// ---------------------------------------------------------------------------

#define T_TOK   2048
#define HDIM    1024
#define FFDIM   512
#define NEXP    16
#define TOPK    4
#define MTILE   32

#define XS_LD   (HDIM + 4)    // padded LDS row stride (floats) -> conflict-free
#define ACT_LD  (FFDIM + 4)

typedef __attribute__((ext_vector_type(2))) float v2f;
typedef __attribute__((ext_vector_type(8))) float v8f;

__device__ __forceinline__ v8f wmma_f32_4(v2f a, v2f b, v8f c) {
    return __builtin_amdgcn_wmma_f32_16x16x4_f32(
        /*neg_a=*/false, a, /*neg_b=*/false, b,
        /*c_mod=*/(short)0, c, /*reuse_a=*/false, /*reuse_b=*/false);
}

// ---------------------------------------------------------------------------
// Kernel 1: gating. One wave32 per token. sigmoid(x @ gate_w.T), top-4 with
// bias (strict '>' argmax == lowest-index tie-break, matches lax.top_k),
// normalized weights; counting-sort (token,slot) pair ids per expert.
// ---------------------------------------------------------------------------
__global__ __launch_bounds__(256)
void afmoe_gate_kernel(const float* __restrict__ x,
                       const float* __restrict__ gw,
                       const float* __restrict__ bias,
                       int*   __restrict__ cnt,      // [NEXP]
                       int*   __restrict__ rows,     // [NEXP][T_TOK] pair ids
                       float* __restrict__ pw)       // [T_TOK*TOPK] weights
{
    const int wv   = threadIdx.x >> 5;
    const int lane = threadIdx.x & 31;
    const int t    = blockIdx.x * 8 + wv;
    if (t >= T_TOK) return;

    float s[NEXP];
    const float* xrow = x + (size_t)t * HDIM;
#pragma unroll 1
    for (int e = 0; e < NEXP; ++e) {
        const float* grow = gw + (size_t)e * HDIM;
        float p = 0.f;
        for (int h = lane; h < HDIM; h += 32) p += xrow[h] * grow[h];
#pragma unroll
        for (int off = 16; off; off >>= 1) p += __shfl_xor(p, off, 32);
        s[e] = 1.f / (1.f + __expf(-p));
    }

    if (lane == 0) {
        int   ids[TOPK];
        float wsel[TOPK];
        bool  used[NEXP] = {};
        float sum = 0.f;
#pragma unroll
        for (int k = 0; k < TOPK; ++k) {
            int best = 0; float bv = -1e30f;
            for (int e = 0; e < NEXP; ++e) {
                float v = s[e] + bias[e];
                if (!used[e] && v > bv) { bv = v; best = e; }
            }
            used[best] = true;
            ids[k]  = best;
            wsel[k] = s[best];
            sum    += s[best];
        }
        float inv = 1.f / fmaxf(sum, 1e-20f);
#pragma unroll
        for (int k = 0; k < TOPK; ++k) {
            int e   = ids[k];
            int pos = atomicAdd(&cnt[e], 1);
            rows[e * T_TOK + pos] = t * TOPK + k;
            pw[t * TOPK + k] = wsel[k] * inv;
        }
    }
}

// ---------------------------------------------------------------------------
// Kernel 2: expert (and shared-expert) MLP via WMMA, 32 rows per block.
// Gather uses GLOBAL_LOAD_ASYNC_TO_LDS_B128 (gfx1250 async path, ASYNCcnt).
// Stage 1: gu tiles -> act = silu(g)*u buffered in registers, then rewritten
//          over the dead xs LDS region. Stage 2: y = act @ w2^T.
// ---------------------------------------------------------------------------
__global__ __launch_bounds__(256)
void afmoe_expert_kernel(const float* __restrict__ x,
                         const float* __restrict__ w1,   // [E][2FF][H]
                         const float* __restrict__ w2,   // [E][H][FF]
                         const float* __restrict__ sw1,  // [2FF][H]
                         const float* __restrict__ sw2,  // [H][FF]
                         const int*   __restrict__ cnt,
                         const int*   __restrict__ rows,
                         const float* __restrict__ pw,
                         float* __restrict__ pair_y,     // [T_TOK*TOPK][H]
                         float* __restrict__ out)        // [T_TOK][H]
{
    const int TILES = T_TOK / MTILE;  // 64 max tiles per expert
    const int b = blockIdx.x;
    const bool is_shared = (b >= NEXP * TILES);

    int e = 0, tile;
    const float *W1, *W2;
    int nv;
    if (!is_shared) {
        e    = b / TILES;
        tile = b % TILES;
        W1   = w1 + (size_t)e * (2 * FFDIM) * HDIM;
        W2   = w2 + (size_t)e * HDIM * FFDIM;
        nv   = cnt[e] - tile * MTILE;
        if (nv <= 0) return;            // uniform exit: empty tile
        if (nv > MTILE) nv = MTILE;
    } else {
        tile = b - NEXP * TILES;
        W1   = sw1;
        W2   = sw2;
        nv   = MTILE;
    }

    // xs (32 x XS_LD floats, ~128.5 KB) is reused as act (32 x ACT_LD) after
    // stage 1: xs is dead once all waves finish their WMMA K-loops.
    __shared__ float smem[MTILE * XS_LD];
    __shared__ int   rpair[MTILE];
    __shared__ int   rtok[MTILE];
    __shared__ float rw[MTILE];

    float* xs   = smem;
    float* actl = smem;

    const int tid = threadIdx.x;
    if (tid < MTILE) {
        int p = -1, tok = 0; float w = 0.f;
        if (!is_shared) {
            if (tid < nv) {
                p   = rows[e * T_TOK + tile * MTILE + tid];
                w   = pw[p];
                tok = p >> 2;
            }
        } else {
            tok = tile * MTILE + tid;
            p   = tok;
            w   = 1.f;
        }
        rpair[tid] = p; rtok[tid] = tok; rw[tid] = w;
    }
    __syncthreads();

    // ---- gather 32 x rows into LDS via async global->LDS (b128/lane) ----
    // One full row (1024 floats = 256 float4) per iteration across the block:
    // block-uniform validity branch, 512B contiguous per wave per issue.
#pragma unroll 1
    for (int r = 0; r < MTILE; ++r) {
        float4* dst = (float4*)(xs + r * XS_LD) + tid;
        if (rpair[r] >= 0) {
            const float4* src = (const float4*)(x + (size_t)rtok[r] * HDIM) + tid;
            unsigned lds_off = (unsigned)(uintptr_t)(void*)dst;  // low 32b of flat = LDS offset
            asm volatile("global_load_async_to_lds_b128 %0, %1, off"
                         :: "v"(lds_off), "v"((unsigned long long)(uintptr_t)src)
                         : "memory");
        } else {
            *dst = make_float4(0.f, 0.f, 0.f, 0.f);
        }
    }
    asm volatile("s_wait_asynccnt 0x0" ::: "memory");
    __syncthreads();

    const int wv   = tid >> 5;
    const int lane = tid & 31;
    const int half = lane >> 4;     // selects K-pair (A/B) and row half (C)
    const int nn   = lane & 15;     // A row / B col / C col

    const float* aP0 = xs + nn * XS_LD + 2 * half;           // rows 0..15
    const float* aP1 = xs + (nn + 16) * XS_LD + 2 * half;    // rows 16..31

    // ---- Stage 1: gu = x_tile @ W1^T ; act = silu(g)*u (reg-buffered) ----
    float actv[4][16];   // [f-tile][C elem: 0..7 rows 0..15, 8..15 rows 16..31]
#pragma unroll
    for (int ft = 0; ft < 4; ++ft) {
        const int f0 = (wv * 4 + ft) * 16;                   // [0, FFDIM)
        v8f cg0 = {}, cu0 = {}, cg1 = {}, cu1 = {};
        const float* bg = W1 + (size_t)(f0 + nn) * HDIM + 2 * half;
        const float* bu = W1 + (size_t)(FFDIM + f0 + nn) * HDIM + 2 * half;
#pragma unroll 4
        for (int k = 0; k < HDIM; k += 4) {
            v2f a0 = *(const v2f*)(aP0 + k);
            v2f a1 = *(const v2f*)(aP1 + k);
            v2f vg = *(const v2f*)(bg + k);
            v2f vu = *(const v2f*)(bu + k);
            cg0 = wmma_f32_4(a0, vg, cg0);
            cu0 = wmma_f32_4(a0, vu, cu0);
            cg1 = wmma_f32_4(a1, vg, cg1);
            cu1 = wmma_f32_4(a1, vu, cu1);
        }
#pragma unroll
        for (int i = 0; i < 8; ++i) {
            float g0 = cg0[i], u0 = cu0[i];
            float g1 = cg1[i], u1 = cu1[i];
            actv[ft][i]     = (g0 / (1.f + __expf(-g0))) * u0;
            actv[ft][8 + i] = (g1 / (1.f + __expf(-g1))) * u1;
        }
    }
    __syncthreads();   // all waves done reading xs -> safe to overwrite

    // ---- spill act registers into (reused) LDS ----
#pragma unroll
    for (int ft = 0; ft < 4; ++ft) {
        const int f0 = (wv * 4 + ft) * 16;
#pragma unroll
        for (int i = 0; i < 8; ++i) {
            int r0 = i + 8 * half;
            actl[r0 * ACT_LD + f0 + nn]        = actv[ft][i];
            actl[(16 + r0) * ACT_LD + f0 + nn] = actv[ft][8 + i];
        }
    }
    __syncthreads();

    // ---- Stage 2: y = act @ W2^T ; scatter scaled rows ----
    const float* aQ0 = actl + nn * ACT_LD + 2 * half;
    const float* aQ1 = actl + (nn + 16) * ACT_LD + 2 * half;
#pragma unroll 1
    for (int nt = 0; nt < 8; ++nt) {
        const int n0 = (wv * 8 + nt) * 16;                   // [0, HDIM)
        v8f c0 = {}, c1 = {};
        const float* bp = W2 + (size_t)(n0 + nn) * FFDIM + 2 * half;
#pragma unroll 4
        for (int k = 0; k < FFDIM; k += 4) {
            v2f a0 = *(const v2f*)(aQ0 + k);
            v2f a1 = *(const v2f*)(aQ1 + k);
            v2f vb = *(const v2f*)(bp + k);
            c0 = wmma_f32_4(a0, vb, c0);
            c1 = wmma_f32_4(a1, vb, c1);
        }
#pragma unroll
        for (int i = 0; i < 8; ++i) {
            int r0 = i + 8 * half;
            int r1 = r0 + 16;
            int p0 = rpair[r0];
            int p1 = rpair[r1];
            if (p0 >= 0) {
                float val = c0[i] * rw[r0];                  // ROUTE_SCALE == 1
                if (!is_shared) pair_y[(size_t)p0 * HDIM + n0 + nn] = val;
                else            out[(size_t)p0 * HDIM + n0 + nn]    = val;
            }
            if (p1 >= 0) {
                float val = c1[i] * rw[r1];
                if (!is_shared) pair_y[(size_t)p1 * HDIM + n0 + nn] = val;
                else            out[(size_t)p1 * HDIM + n0 + nn]    = val;
            }
        }
    }
}

// ---------------------------------------------------------------------------
// Kernel 3: out[t][h] = shared_out[t][h] + sum_k pair_y[t*4+k][h]
// (fixed k order -> bit-deterministic; no float atomics anywhere)
// ---------------------------------------------------------------------------
__global__ __launch_bounds__(256)
void afmoe_combine_kernel(const float* __restrict__ pair_y,
                          float* __restrict__ out)
{
    int idx = blockIdx.x * 256 + threadIdx.x;               // over T_TOK*HDIM
    int t = idx / HDIM;
    int h = idx - t * HDIM;
    float acc = out[idx];
#pragma unroll
    for (int k = 0; k < TOPK; ++k)
        acc += pair_y[(size_t)(t * TOPK + k) * HDIM + h];
    out[idx] = acc;
}

// ---------------------------------------------------------------------------
extern "C" void kernel_launch(void* const* d_in, const int* in_sizes, int n_in,
                              void* d_out, int out_size, void* d_ws, size_t ws_size,
                              hipStream_t stream) {
    (void)in_sizes; (void)n_in; (void)out_size; (void)ws_size;

    const float* x    = (const float*)d_in[0];   // [T,H]
    const float* gw   = (const float*)d_in[1];   // [E,H]
    const float* bias = (const float*)d_in[2];   // [E]
    const float* w1   = (const float*)d_in[3];   // [E,2FF,H]
    const float* w2   = (const float*)d_in[4];   // [E,H,FF]
    const float* sw1  = (const float*)d_in[5];   // [2FF,H]
    const float* sw2  = (const float*)d_in[6];   // [H,FF]
    float* out = (float*)d_out;

    char* ws = (char*)d_ws;
    int*   cnt    = (int*)(ws);                                   // E ints
    int*   rows   = (int*)(ws + 1024);                            // E*T ints
    float* pw     = (float*)(ws + 1024 + NEXP * T_TOK * 4);       // T*4 floats
    float* pair_y = (float*)(ws + 1024 + NEXP * T_TOK * 4 + T_TOK * TOPK * 4); // 32 MB

    hipMemsetAsync(cnt, 0, NEXP * sizeof(int), stream);

    afmoe_gate_kernel<<<T_TOK / 8, 256, 0, stream>>>(x, gw, bias, cnt, rows, pw);

    const int TILES = T_TOK / MTILE;
    afmoe_expert_kernel<<<(NEXP + 1) * TILES, 256, 0, stream>>>(
        x, w1, w2, sw1, sw2, cnt, rows, pw, pair_y, out);

    afmoe_combine_kernel<<<(T_TOK * HDIM) / 256, 256, 0, stream>>>(pair_y, out);
}